// S4DKernel_31396210933767
// MI455X (gfx1250) — compile-verified
//
#include <hip/hip_runtime.h>
#include <math.h>

// S4D kernel generation for MI455X (gfx1250), wave32 + WMMA fp32.
//
// K[h, t] = 2*Re( sum_n C_eff[h,n] * exp(dtA[h,n])^t ),  t in [0, 4096)
// Factor t = 64*q + r:  z^t = z^(64q) * z^r
//   U[n][q] = 2*C_eff[n]*z_n^(64q)   (32x64 complex)
//   V[n][r] = z_n^r                  (32x64 complex)
//   K[q][r] = Re( U^T V )  -> real contraction over 64 (re/-im interleave)
// Contraction done with V_WMMA_F32_16X16X4_F32 (full fp32 precision).

typedef __attribute__((ext_vector_type(2))) float v2f;
typedef __attribute__((ext_vector_type(8))) float v8f;

#define NHALF 32
#define L_TOT 4096
#define LDSTRIDE 80   // padded row stride (64 data + 16 pad) -> halves hit disjoint banks

__global__ __launch_bounds__(256) void s4d_vand_wmma_kernel(
    const float* __restrict__ log_dt,      // (H,)
    const float* __restrict__ C,           // (H, 32, 2) re/im
    const float* __restrict__ log_A_real,  // (H, 32)
    const float* __restrict__ A_imag,      // (H, 32)
    float* __restrict__ out)               // (H, 4096)
{
    // Folded operand matrices in LDS:
    //  Af[k][q]: k=2n -> +Re U[n][q], k=2n+1 -> -Im U[n][q]
    //  Bf[k][r]: k=2n -> +Re V[n][r], k=2n+1 -> +Im V[n][r]
    __shared__ float Af[64 * LDSTRIDE];
    __shared__ float Bf[64 * LDSTRIDE];

    const int h   = blockIdx.x;
    const int tid = threadIdx.x;

    const float dt = expf(log_dt[h]);

    // ---------------- Phase 1: fill U (Af) and V (Bf) ----------------
    // 32 modes x 64 columns = 2048 complex entries each; 8 per thread.
    #pragma unroll
    for (int i = 0; i < 8; ++i) {
        const int idx = i * 256 + tid;   // 0..2047
        const int n   = idx >> 6;        // mode 0..31
        const int q   = idx & 63;        // column 0..63

        const float Ar   = -expf(log_A_real[h * NHALF + n]);
        const float Ai   = A_imag[h * NHALF + n];
        const float dtAr = dt * Ar;
        const float dtAi = dt * Ai;

        // C_eff = C * (exp(dtA) - 1) / A   (folding in the final factor 2)
        const float e1 = expf(dtAr);
        float s1, c1;
        sincosf(dtAi, &s1, &c1);
        const float numr = e1 * c1 - 1.0f;
        const float numi = e1 * s1;
        const float inv  = 1.0f / (Ar * Ar + Ai * Ai);
        const float Fr   = (numr * Ar + numi * Ai) * inv;
        const float Fi   = (numi * Ar - numr * Ai) * inv;
        const float Cr   = C[(h * NHALF + n) * 2 + 0];
        const float Ci   = C[(h * NHALF + n) * 2 + 1];
        const float Er   = 2.0f * (Cr * Fr - Ci * Fi);
        const float Ei   = 2.0f * (Cr * Fi + Ci * Fr);

        // U[n][q] = C_eff * exp(dtA * 64q)
        const float t64 = 64.0f * (float)q;
        const float mq  = expf(dtAr * t64);
        float sq, cq;
        sincosf(dtAi * t64, &sq, &cq);   // up to ~4e4 rad: accurate range reduction
        const float Zr = mq * cq, Zi = mq * sq;
        Af[(2 * n)     * LDSTRIDE + q] =  Er * Zr - Ei * Zi;   // +Re U
        Af[(2 * n + 1) * LDSTRIDE + q] = -(Er * Zi + Ei * Zr); // -Im U

        // V[n][r] = exp(dtA * r), reuse r == q
        const float fr = (float)q;
        const float mr = expf(dtAr * fr);
        float sr, cr;
        sincosf(dtAi * fr, &sr, &cr);
        Bf[(2 * n)     * LDSTRIDE + q] = mr * cr;              // +Re V
        Bf[(2 * n + 1) * LDSTRIDE + q] = mr * sr;              // +Im V
    }
    __syncthreads();

    // ---------------- Phase 2: WMMA contraction ----------------
    // 64x64 output = 16 tiles of 16x16; 8 waves -> 2 tiles/wave.
    const int lane  = tid & 31;       // wave32
    const int wave  = tid >> 5;       // 0..7
    const int m     = lane & 15;      // M (A) / N (B,D) index within tile
    const int khalf = lane >> 4;      // 0: K=0,1   1: K=2,3  (per-instruction)

    #pragma unroll
    for (int tt = 0; tt < 2; ++tt) {
        const int tile  = wave + tt * 8;       // 0..15
        const int qbase = (tile >> 2) * 16;    // coarse-time tile base
        const int rbase = (tile & 3) * 16;     // fine-time tile base

        v8f acc = {0.f, 0.f, 0.f, 0.f, 0.f, 0.f, 0.f, 0.f};

        #pragma unroll
        for (int kk = 0; kk < 16; ++kk) {      // K=64 in chunks of 4
            const int kb = kk * 4 + 2 * khalf;
            v2f a, b;
            // A 16x4 fp32 layout: lanes 0-15 hold K=0(v0),1(v1); lanes 16-31 K=2,3
            a.x = Af[(kb    ) * LDSTRIDE + qbase + m];
            a.y = Af[(kb + 1) * LDSTRIDE + qbase + m];
            // B 4x16 fp32: rows striped across lanes (mirror of A)
            b.x = Bf[(kb    ) * LDSTRIDE + rbase + m];
            b.y = Bf[(kb + 1) * LDSTRIDE + rbase + m];
            acc = __builtin_amdgcn_wmma_f32_16x16x4_f32(
                /*neg_a=*/false, a, /*neg_b=*/false, b,
                /*c_mod=*/(short)0, acc,
                /*reuse_a=*/false, /*reuse_b=*/false);
        }

        // D layout: vgpr v -> row M = v + 8*khalf, col N = lane&15
        float* dst = out + (size_t)h * L_TOT;
        #pragma unroll
        for (int v = 0; v < 8; ++v) {
            const int row = qbase + v + 8 * khalf;          // q index
            dst[row * 64 + rbase + m] = acc[v];             // t = 64q + r
        }
    }
}

extern "C" void kernel_launch(void* const* d_in, const int* in_sizes, int n_in,
                              void* d_out, int out_size, void* d_ws, size_t ws_size,
                              hipStream_t stream) {
    // setup_inputs order: L (int scalar), log_dt, C, log_A_real, A_imag
    const float* log_dt     = (const float*)d_in[1];
    const float* C          = (const float*)d_in[2];
    const float* log_A_real = (const float*)d_in[3];
    const float* A_imag     = (const float*)d_in[4];
    float* out = (float*)d_out;

    const int H = in_sizes[1];  // 1024 channels; L fixed at 4096 (compile-time factorization)

    s4d_vand_wmma_kernel<<<H, 256, 0, stream>>>(log_dt, C, log_A_real, A_imag, out);
    (void)in_sizes; (void)n_in; (void)out_size; (void)d_ws; (void)ws_size;
}